// Augmentation_46136538694218
// MI455X (gfx1250) — compile-verified
//
#include <hip/hip_runtime.h>

// MI455X (gfx1250) affine-grid + bilinear grid_sample (border, align_corners=False).
// N=32, C=3, H=W=512.  Memory-bound: ~134MB gathered reads (fits 192MB L2),
// ~134MB streamed writes (non-temporal to preserve L2 residency of inputs).
// The affine grid itself is computed with V_WMMA_F32_16X16X4_F32:
//   D[2 x 16pix] = theta[2x3 padded to 16x4] * base[4 x 16pix]

typedef float v2f __attribute__((ext_vector_type(2)));
typedef float v8f __attribute__((ext_vector_type(8)));

#define IMG_HW 262144    // 512*512
#define IMG_W  512

__global__ __launch_bounds__(256) void Augmentation_46136538694218_kernel(
    const float* __restrict__ input_g,   // [32,3,512,512]
    const float* __restrict__ label_g,   // [32,1,512,512]
    const float* __restrict__ theta,     // [32,2,3]
    float* __restrict__ out_img,         // [32,3,512,512]
    float* __restrict__ out_lbl)         // [32,1,512,512] (bool as 0/1 float)
{
    // 1024 blocks per sample (262144 px / 256 threads) -> n is block-uniform (scalar).
    const int n   = blockIdx.x >> 10;
    const int rem = ((blockIdx.x & 1023) << 8) | threadIdx.x;   // pixel index in sample
    const int h   = rem >> 9;
    const int w   = rem & 511;
    const int lane = threadIdx.x & 31;
    const bool lo  = lane < 16;

    // theta row-major: t00 t01 t02 / t10 t11 t12  (scalarized loads, n uniform)
    const float* tp = theta + n * 6;
    const float t00 = tp[0], t01 = tp[1], t02 = tp[2];
    const float t10 = tp[3], t11 = tp[4], t12 = tp[5];

    const float inv = 2.0f / 512.0f;
    const float gy  = (h + 0.5f) * inv - 1.0f;     // wave-uniform (512 % 32 == 0)

    // ---- A matrix (16x4 f32): lanes 0-15 hold K0/K1, lanes 16-31 hold K2/K3 ----
    // Row M=0 = [t00 t01 t02 0]  (x),  Row M=1 = [t10 t11 t12 0]  (y), rows 2..15 = 0.
    float a0 = 0.0f, a1 = 0.0f;
    if      (lane == 0)  { a0 = t00; a1 = t01; }
    else if (lane == 1)  { a0 = t10; a1 = t11; }
    else if (lane == 16) { a0 = t02; }           // K=2 for M=0
    else if (lane == 17) { a0 = t12; }           // K=2 for M=1
    v2f A; A.x = a0; A.y = a1;

    // ---- B matrices (4x16 f32): VGPR0 = {K0 | K2}, VGPR1 = {K1 | K3} ----
    // Column N = pixel; rows: K0=gx, K1=gy, K2=1, K3=0.
    // Wave covers 32 consecutive w; B1 -> pixels [wbase..wbase+15], B2 -> [+16..+31].
    v2f B1, B2;
    B1.x = lo ? ((w + 0.5f) * inv - 1.0f)        : 1.0f;
    B1.y = lo ? gy                               : 0.0f;
    B2.x = lo ? ((w + 16 + 0.5f) * inv - 1.0f)   : 1.0f;
    B2.y = B1.y;

    v8f Cz = {};
    // (neg_a, A, neg_b, B, c_mod, C, reuse_a, reuse_b)
    v8f D1 = __builtin_amdgcn_wmma_f32_16x16x4_f32(false, A, false, B1, (short)0, Cz, false, false);
    v8f D2 = __builtin_amdgcn_wmma_f32_16x16x4_f32(false, A, false, B2, (short)0, Cz, false, false);

    // D[0][N]=x, D[1][N]=y live in lanes 0-15. Upper half-wave pulls its pixel's
    // coords from D2 of lane (lane-16).
    const float x2 = __shfl(D2[0], lane & 15);
    const float y2 = __shfl(D2[1], lane & 15);
    const float gxn = lo ? D1[0] : x2;
    const float gyn = lo ? D1[1] : y2;

    // ---- bilinear sample, padding_mode='border', align_corners=False ----
    const float fx = fminf(fmaxf((gxn + 1.0f) * 256.0f - 0.5f, 0.0f), 511.0f);
    const float fy = fminf(fmaxf((gyn + 1.0f) * 256.0f - 0.5f, 0.0f), 511.0f);
    const float x0f = floorf(fx), y0f = floorf(fy);
    const float wx1 = fx - x0f,  wy1 = fy - y0f;
    const float wx0 = 1.0f - wx1, wy0 = 1.0f - wy1;
    const int x0 = (int)x0f,       y0 = (int)y0f;
    const int x1 = min(x0 + 1, 511), y1 = min(y0 + 1, 511);
    const float w00 = wy0 * wx0, w01 = wy0 * wx1;
    const float w10 = wy1 * wx0, w11 = wy1 * wx1;

    const int i00 = (y0 << 9) + x0, i01 = (y0 << 9) + x1;
    const int i10 = (y1 << 9) + x0, i11 = (y1 << 9) + x1;

    // Image: 3 channels, gathers hit L2 (whole 134MB working set resident).
    const float* in_n = input_g + (size_t)n * (3 * IMG_HW);
    float*       o_n  = out_img + (size_t)n * (3 * IMG_HW);
#pragma unroll
    for (int c = 0; c < 3; ++c) {
        const float* p = in_n + c * IMG_HW;
        float v = fmaf(p[i00], w00,
                  fmaf(p[i01], w01,
                  fmaf(p[i10], w10, p[i11] * w11)));
        __builtin_nontemporal_store(v, o_n + c * IMG_HW + rem);   // stream, keep L2 for inputs
    }

    // Label: same coords/weights, then threshold.
    const float* q = label_g + (size_t)n * IMG_HW;
    float lv = fmaf(q[i00], w00,
               fmaf(q[i01], w01,
               fmaf(q[i10], w10, q[i11] * w11)));
    __builtin_nontemporal_store(lv > 0.5f ? 1.0f : 0.0f,
                                out_lbl + (size_t)n * IMG_HW + rem);
}

extern "C" void kernel_launch(void* const* d_in, const int* in_sizes, int n_in,
                              void* d_out, int out_size, void* d_ws, size_t ws_size,
                              hipStream_t stream) {
    (void)in_sizes; (void)n_in; (void)out_size; (void)d_ws; (void)ws_size;
    const float* input_g = (const float*)d_in[0];   // 32*3*512*512
    const float* label_g = (const float*)d_in[1];   // 32*1*512*512
    const float* theta   = (const float*)d_in[2];   // 32*2*3
    float* out_img = (float*)d_out;                 // first output, flat
    float* out_lbl = out_img + (size_t)32 * 3 * IMG_HW;  // second output follows

    dim3 block(256);                 // 8 wave32 per block
    dim3 grid(32 * 1024);            // one thread per output pixel, exact cover
    Augmentation_46136538694218_kernel<<<grid, block, 0, stream>>>(
        input_g, label_g, theta, out_img, out_lbl);
}